// GINConv_57312043597895
// MI455X (gfx1250) — compile-verified
//
#include <hip/hip_runtime.h>

typedef __attribute__((ext_vector_type(2))) float v2f;
typedef __attribute__((ext_vector_type(8))) float v8f;

#define N_NODES 50000
#define N_EDGES 800000
#define D 128

// ---------------------------------------------------------------------------
// Kernel 1: zero the neighbor-accumulator in workspace (float4 stores)
// ---------------------------------------------------------------------------
__global__ void gin_zero_kernel(float4* __restrict__ p, int n4) {
    int i = blockIdx.x * blockDim.x + threadIdx.x;
    if (i < n4) {
        p[i] = make_float4(0.f, 0.f, 0.f, 0.f);
    }
}

// ---------------------------------------------------------------------------
// Kernel 2: edge scatter-add.  One wave (32 lanes) per edge; each lane moves
// one float4 of the 128-float feature row.  feat (25.6 MB) and neigh (25.6 MB)
// are L2-resident on MI455X (192 MB L2), so the random gather + atomics stay
// on-chip.  Lanes of a wave hit distinct dwords of one row -> no intra-wave
// atomic conflicts.
// ---------------------------------------------------------------------------
__global__ void gin_scatter_kernel(const float* __restrict__ feat,
                                   const int* __restrict__ src,
                                   const int* __restrict__ dst,
                                   float* __restrict__ neigh) {
    int edge = (blockIdx.x * blockDim.x + threadIdx.x) >> 5;
    int lane = threadIdx.x & 31;
    if (edge >= N_EDGES) return;
    int s = src[edge];
    int t = dst[edge];
    const float4* frow = (const float4*)(feat + (size_t)s * D);
    float4 v = frow[lane];
    float* nrow = neigh + (size_t)t * D + lane * 4;
    atomicAdd(nrow + 0, v.x);
    atomicAdd(nrow + 1, v.y);
    atomicAdd(nrow + 2, v.z);
    atomicAdd(nrow + 3, v.w);
}

// ---------------------------------------------------------------------------
// Kernel 3: out = ((1+eps)*feat + neigh) @ W + b  via V_WMMA_F32_16X16X4_F32.
// One wave computes one 16x16 output tile; 8 waves per block share one M-tile
// (same A rows -> L2/WGP$ reuse), wave w owns N-tile w.  K loop: 32 WMMA ops.
//
// Operand layouts (CDNA5 ISA 7.12.2):
//   A 16x4 f32 : lane(l,h)  VGPR0 = A[l&15][2h],  VGPR1 = A[l&15][2h+1]
//   B 4x16 f32 : lane(l,h)  VGPR0 = B[2h][l&15],  VGPR1 = B[2h+1][l&15]
//   C/D 16x16  : VGPR v     -> row (v + 8h), col (l&15)
// ---------------------------------------------------------------------------
__global__ void gin_gemm_kernel(const float* __restrict__ feat,
                                const float* __restrict__ neigh,
                                const float* __restrict__ W,
                                const float* __restrict__ bias,
                                const float* __restrict__ eps,
                                float* __restrict__ out) {
    const int lane = threadIdx.x & 31;
    const int wave = threadIdx.x >> 5;     // 0..7  -> N tile index
    const int half = lane >> 4;            // 0 or 1
    const int l15  = lane & 15;

    const int row0 = blockIdx.x * 16;      // M tile (50000 = 3125 * 16, exact)
    const int n0   = wave * 16;            // N tile

    const float scale = 1.0f + eps[0];

    const float* arow_f = feat  + (size_t)(row0 + l15) * D;
    const float* arow_n = neigh + (size_t)(row0 + l15) * D;

    v8f acc = {};
    #pragma unroll 4
    for (int k0 = 0; k0 < D; k0 += 4) {
        const int ka = k0 + half * 2;
        v2f a;
        a.x = scale * arow_f[ka]     + arow_n[ka];
        a.y = scale * arow_f[ka + 1] + arow_n[ka + 1];
        v2f bm;
        bm.x = W[(size_t)(ka)     * D + n0 + l15];
        bm.y = W[(size_t)(ka + 1) * D + n0 + l15];
        acc = __builtin_amdgcn_wmma_f32_16x16x4_f32(
            /*neg_a=*/false, a, /*neg_b=*/false, bm,
            /*c_mod=*/(short)0, acc, /*reuse_a=*/false, /*reuse_b=*/false);
    }

    const float bv = bias[n0 + l15];
    float* orow = out + (size_t)(row0 + half * 8) * D + n0 + l15;
    #pragma unroll
    for (int v = 0; v < 8; ++v) {
        orow[(size_t)v * D] = acc[v] + bv;
    }
}

// ---------------------------------------------------------------------------
extern "C" void kernel_launch(void* const* d_in, const int* in_sizes, int n_in,
                              void* d_out, int out_size, void* d_ws, size_t ws_size,
                              hipStream_t stream) {
    const float* feat = (const float*)d_in[0];
    const float* W    = (const float*)d_in[1];
    const float* b    = (const float*)d_in[2];
    const float* eps  = (const float*)d_in[3];
    const int*   src  = (const int*)d_in[4];
    const int*   dst  = (const int*)d_in[5];
    float*       out  = (float*)d_out;
    float*       neigh = (float*)d_ws;     // N_NODES * D floats = 25.6 MB

    // 1) neigh = 0
    const int n4 = (N_NODES * D) / 4;                       // 1.6M float4
    gin_zero_kernel<<<(n4 + 255) / 256, 256, 0, stream>>>((float4*)neigh, n4);

    // 2) scatter-add: one wave per edge
    const long long threads = (long long)N_EDGES * 32;      // 25.6M threads
    gin_scatter_kernel<<<(int)((threads + 255) / 256), 256, 0, stream>>>(
        feat, src, dst, neigh);

    // 3) fused (1+eps)*feat + neigh, GEMM with W, + bias  (WMMA f32)
    gin_gemm_kernel<<<N_NODES / 16, 256, 0, stream>>>(
        feat, neigh, W, b, eps, out);
}